// LogEig_36661840838888
// MI455X (gfx1250) — compile-verified
//
#include <hip/hip_runtime.h>
#include <math.h>

// LogEig via GEMM-only inverse-scaling-and-squaring on CDNA5 WMMA (fp32 matrix pipe).
//   log P = ln(tr P) * I + 2^K * log( (P/tr P)^(1/2^K) ),
// square roots by coupled Newton-Schulz (all 64x64 matmuls), residual log by
// degree-8 Horner series. All iterates are symmetric polynomials in A, which
// lets both WMMA operands be fetched as contiguous LDS rows (ds_load_b64).

typedef __attribute__((ext_vector_type(2))) float v2f;
typedef __attribute__((ext_vector_type(8))) float v8f;

#define NMAT   64
#define STRIDE 68              // padded row stride (floats): conflict-free banks
#define MATF   (NMAT * STRIDE) // 4352 floats per LDS matrix buffer
#define NSQRT  5               // number of square roots (2^5 = 32)

// dst = alpha * (A * B) + diag * I   (64x64, all row-major fp32 in LDS,
// B must be symmetric: its columns are fetched as rows).
// 256 threads = 8 waves; wave w: rows [ (w>>1)*16 ), cols [ (w&1)*32, +32 ).
__device__ __forceinline__ void mm64(float* __restrict__ dst,
                                     const float* __restrict__ A,
                                     const float* __restrict__ Bsym,
                                     float alpha, float diag, int tid)
{
    const int lane = tid & 31;
    const int wave = tid >> 5;
    const int m0 = (wave >> 1) << 4;      // 0,16,32,48
    const int n0 = (wave & 1) << 5;       // 0 or 32; this wave does n0 and n0+16
    const int r  = lane & 15;
    const int kh = (lane < 16) ? 0 : 2;   // f32 WMMA A/B frag: K split 0/1 vs 2/3

    const float* arow  = A    + (m0 + r) * STRIDE;
    const float* b0row = Bsym + (n0 + r) * STRIDE;        // col n == row n (symmetric)
    const float* b1row = Bsym + (n0 + 16 + r) * STRIDE;

    v8f acc0 = {0.f,0.f,0.f,0.f,0.f,0.f,0.f,0.f};
    v8f acc1 = {0.f,0.f,0.f,0.f,0.f,0.f,0.f,0.f};
    #pragma unroll
    for (int k = 0; k < NMAT; k += 4) {
        v2f a  = *(const v2f*)(arow  + k + kh);
        v2f b0 = *(const v2f*)(b0row + k + kh);
        v2f b1 = *(const v2f*)(b1row + k + kh);
        acc0 = __builtin_amdgcn_wmma_f32_16x16x4_f32(false, a, false, b0,
                                                     (short)0, acc0, false, false);
        acc1 = __builtin_amdgcn_wmma_f32_16x16x4_f32(false, a, false, b1,
                                                     (short)0, acc1, false, false);
    }

    // C/D layout: VGPR i -> row m0 + i + (lane<16?0:8), col = n + (lane&15)
    const int rofs = (lane < 16) ? 0 : 8;
    const int c0 = n0 + r;
    const int c1 = n0 + 16 + r;
    #pragma unroll
    for (int i = 0; i < 8; ++i) {
        int row = m0 + rofs + i;
        dst[row * STRIDE + c0] = alpha * acc0[i] + ((row == c0) ? diag : 0.f);
        dst[row * STRIDE + c1] = alpha * acc1[i] + ((row == c1) ? diag : 0.f);
    }
}

__global__ __launch_bounds__(256)
void logeig_kernel(const float* __restrict__ P, float* __restrict__ out, int nmat)
{
    __shared__ float smem[4 * MATF + 16];
    float* buf0 = smem;
    float* red  = smem + 4 * MATF;

    const int mat = blockIdx.x;
    if (mat >= nmat) return;
    const int tid = threadIdx.x;
    const float*  Pg = P   + (size_t)mat * (NMAT * NMAT);
    float*        Og = out + (size_t)mat * (NMAT * NMAT);

    int bY = 0, bZ = 1, bT = 2, bU = 3;   // rotating LDS matrix buffers
    #define BUF(i) (buf0 + (i) * MATF)

    // ---- load P (coalesced), accumulate trace via LDS float atomic ----
    if (tid == 0) red[0] = 0.f;
    __syncthreads();
    #pragma unroll
    for (int i = 0; i < 16; ++i) {
        int idx = (i << 8) + tid;               // 0..4095, consecutive per wave
        float v = Pg[idx];
        int row = idx >> 6, col = idx & 63;
        BUF(bY)[row * STRIDE + col] = v;
        if (row == col) atomicAdd(red, v);      // ds_add_f32
    }
    __syncthreads();
    const float tr    = red[0];
    const float invtr = 1.0f / tr;

    // ---- A = P / tr  (eigenvalues in (0,1]) ----
    #pragma unroll
    for (int i = 0; i < 16; ++i) {
        int idx = (i << 8) + tid;
        int row = idx >> 6, col = idx & 63;
        BUF(bY)[row * STRIDE + col] *= invtr;
    }
    __syncthreads();

    // ---- K square roots via coupled Newton-Schulz ----
    const int iters[NSQRT] = {16, 11, 8, 6, 5};   // heavier early (worst cond)
    for (int s = 0; s < NSQRT; ++s) {
        // Z = I
        #pragma unroll
        for (int i = 0; i < 16; ++i) {
            int idx = (i << 8) + tid;
            int row = idx >> 6, col = idx & 63;
            BUF(bZ)[row * STRIDE + col] = (row == col) ? 1.f : 0.f;
        }
        __syncthreads();
        for (int it = 0; it < iters[s]; ++it) {
            mm64(BUF(bT), BUF(bZ), BUF(bY), -0.5f, 1.5f, tid);  // T = 1.5I - 0.5 Z*Y
            __syncthreads();
            mm64(BUF(bU), BUF(bY), BUF(bT), 1.0f, 0.0f, tid);   // U = Y*T   (new Y)
            __syncthreads();
            mm64(BUF(bY), BUF(bT), BUF(bZ), 1.0f, 0.0f, tid);   // oldY <- T*Z (new Z)
            __syncthreads();
            int t0 = bY, t1 = bZ, t2 = bT, t3 = bU;
            bY = t3; bZ = t0; bT = t1; bU = t2;                 // rotate buffers
        }
    }

    // ---- log(I+E), E = Y - I, degree-8 series via Horner (6 GEMMs + final) ----
    // q(E) = sum_{j=0..7} c_j E^j,  c_j = (-1)^j/(j+1);  log(I+E) = E * q(E)
    const float c0f = 1.f,        c1f = -0.5f,     c2f = 1.f/3.f, c3f = -0.25f;
    const float c4f = 0.2f,       c5f = -1.f/6.f,  c6f = 1.f/7.f, c7f = -0.125f;
    const int bE = bZ;                     // Z is dead now
    #pragma unroll
    for (int i = 0; i < 16; ++i) {
        int idx = (i << 8) + tid;
        int row = idx >> 6, col = idx & 63;
        float y = BUF(bY)[row * STRIDE + col];
        float e = y - ((row == col) ? 1.f : 0.f);
        BUF(bE)[row * STRIDE + col] = e;
        BUF(bT)[row * STRIDE + col] = c7f * e + ((row == col) ? c6f : 0.f); // S init
    }
    __syncthreads();
    int bS = bT, bS2 = bU;
    const float cs[6] = {c5f, c4f, c3f, c2f, c1f, c0f};
    for (int j = 0; j < 6; ++j) {
        mm64(BUF(bS2), BUF(bS), BUF(bE), 1.0f, cs[j], tid);     // S = S*E + c_j I
        __syncthreads();
        int t = bS; bS = bS2; bS2 = t;
    }
    // out = 2^K * (E * S) + ln(tr) * I   (fused into GEMM epilogue)
    const float scale = (float)(1 << NSQRT);
    mm64(BUF(bS2), BUF(bE), BUF(bS), scale, scale * 0.f + logf(tr), tid);
    __syncthreads();

    // ---- coalesced store ----
    #pragma unroll
    for (int i = 0; i < 16; ++i) {
        int idx = (i << 8) + tid;
        int row = idx >> 6, col = idx & 63;
        Og[idx] = BUF(bS2)[row * STRIDE + col];
    }
    #undef BUF
}

extern "C" void kernel_launch(void* const* d_in, const int* in_sizes, int n_in,
                              void* d_out, int out_size, void* d_ws, size_t ws_size,
                              hipStream_t stream) {
    (void)n_in; (void)out_size; (void)d_ws; (void)ws_size;
    const float* P = (const float*)d_in[0];
    float* out = (float*)d_out;
    const int nmat = in_sizes[0] / (NMAT * NMAT);   // B*H = 8192
    hipLaunchKernelGGL(logeig_kernel, dim3(nmat), dim3(256), 0, stream, P, out, nmat);
}